// BasicBlock_52536039965259
// MI455X (gfx1250) — compile-verified
//
#include <hip/hip_runtime.h>

#define EPSV 1e-5f

typedef __attribute__((ext_vector_type(2))) float v2f;
typedef __attribute__((ext_vector_type(8))) float v8f;

// ---------------------------------------------------------------------------
// One-time weight repack: w[o][i][r][s] (OIHW, 64x64x3x3) ->
// wt[conv][tap][(k>>1)*128 + n*2 + (k&1)]  (k = input channel, n = out channel)
// This is exactly the per-lane order needed for WMMA B fragments, so the main
// kernel can stage a 16KB tap-slab into LDS with coalesced float4 loads.
// ---------------------------------------------------------------------------
__global__ __launch_bounds__(256) void prep_weights_kernel(
    const float* __restrict__ w1, const float* __restrict__ w2,
    float* __restrict__ wt)
{
    int i = blockIdx.x * 256 + threadIdx.x;
    if (i >= 2 * 36864) return;
    const float* w = (i < 36864) ? w1 : w2;
    float* dst = wt + (i < 36864 ? 0 : 36864);
    int j = i % 36864;
    int tap = j / 4096;
    int rem = j % 4096;
    int n = (rem >> 1) & 63;
    int k = ((rem >> 7) << 1) | (rem & 1);
    dst[j] = w[(n * 64 + k) * 9 + tap];
}

// ---------------------------------------------------------------------------
// Fused BasicBlock: one workgroup (8 wave32) per batch image.
//   GEMM view per conv: M=49 spatial (padded to 64), N=64 out-ch, K=64 in-ch,
//   summed over 9 filter taps. V_WMMA_F32_16X16X4_F32 keeps exact fp32.
//   Wave w owns tiles (mt=w>>2, nt=w&3) and (mt+2, nt) -> shared B fragment.
// ---------------------------------------------------------------------------
__global__ __launch_bounds__(256) void fused_basicblock_kernel(
    const float* __restrict__ x,
    const float* __restrict__ wt,      // repacked weights: [2][9][4096]
    const float* __restrict__ g1, const float* __restrict__ b1,
    const float* __restrict__ m1, const float* __restrict__ v1,
    const float* __restrict__ g2, const float* __restrict__ b2,
    const float* __restrict__ m2, const float* __restrict__ v2p,
    const float* __restrict__ mask1, const float* __restrict__ mask2,
    float* __restrict__ out)
{
    __shared__ __align__(16) float xpad[64 * 81];   // padded 9x9 input tile
    __shared__ __align__(16) float ipad[64 * 81];   // padded intermediate tile
    __shared__ __align__(16) float wsh[64 * 64];    // per-tap weight slab

    const int tid  = threadIdx.x;
    const int lane = tid & 31;
    const int wave = tid >> 5;
    const int b    = blockIdx.x;

    const int lo  = lane & 15;
    const int hi  = lane >> 4;       // 0/1: which half of the wave
    const int khi = hi * 2;          // k offset of this lane within a K=4 step

    // Tile geometry for this wave's two output tiles (share column block nn).
    const int nn  = (wave & 3) * 16 + lo;     // out-channel column
    const int mt0 = (wave >> 2);              // 0..1 ; second tile is mt0+2
    int mA0 = mt0 * 16 + lo;        if (mA0 > 48) mA0 = 48;
    int mA1 = (mt0 + 2) * 16 + lo;  if (mA1 > 48) mA1 = 48;
    const int y0 = mA0 / 7, x0c = mA0 % 7;
    const int y1 = mA1 / 7, x1c = mA1 % 7;

    // Zero padded tiles (borders must be zero for the 3x3 halo).
    for (int i = tid; i < 64 * 81; i += 256) { xpad[i] = 0.0f; ipad[i] = 0.0f; }

    // Stage this image's activations into the padded LDS tile.
    const float* xb = x + (size_t)b * 3136;
    for (int i = tid; i < 3136; i += 256) {
        int c = i / 49, m = i % 49;
        xpad[c * 81 + (m / 7 + 1) * 9 + (m % 7) + 1] = xb[i];
    }
    __syncthreads();

    // One conv pass: 9 taps x 16 K-steps of V_WMMA_F32_16X16X4_F32.
    auto conv_pass = [&](const float* src, const float* __restrict__ wslab,
                         v8f& acc0, v8f& acc1) {
        for (int tap = 0; tap < 9; ++tap) {
            __syncthreads();   // previous slab fully consumed
            {
                const float4* s4 = (const float4*)(wslab + tap * 4096);
                float4* d4 = (float4*)wsh;
                #pragma unroll
                for (int i = 0; i < 4; ++i) d4[tid + i * 256] = s4[tid + i * 256];
            }
            if (tap < 8)  // pull next tap slab toward L2/L0 (global_prefetch_b8)
                __builtin_prefetch((const char*)(wslab + (tap + 1) * 4096) + tid * 64, 0, 1);
            __syncthreads();

            const int r = tap / 3, s = tap - r * 3;
            const int so0 = (y0 + r) * 9 + (x0c + s);
            const int so1 = (y1 + r) * 9 + (x1c + s);
            #pragma unroll
            for (int ks = 0; ks < 16; ++ks) {
                const int kl = ks * 4 + khi;
                v2f a0, a1;
                a0[0] = src[kl * 81 + so0];
                a0[1] = src[kl * 81 + 81 + so0];
                a1[0] = src[kl * 81 + so1];
                a1[1] = src[kl * 81 + 81 + so1];
                const v2f bf = *(const v2f*)&wsh[(ks * 2 + hi) * 128 + nn * 2];
                acc0 = __builtin_amdgcn_wmma_f32_16x16x4_f32(
                    false, a0, false, bf, (short)0, acc0, false, false);
                acc1 = __builtin_amdgcn_wmma_f32_16x16x4_f32(
                    false, a1, false, bf, (short)0, acc1, false, false);
            }
        }
    };

    // ---------------- conv1 ----------------
    v8f c1a = {}, c1b = {};
    conv_pass(xpad, wt, c1a, c1b);

    // Epilogue 1: BN1 + ReLU + mask1(batch 0) -> padded intermediate tile.
    const float sc1 = g1[nn] * rsqrtf(v1[nn] + EPSV);
    const float sh1 = b1[nn] - m1[nn] * sc1;
    auto epi1 = [&](const v8f& acc, int mt) {
        #pragma unroll
        for (int j = 0; j < 8; ++j) {
            int m = mt * 16 + j + 8 * hi;
            if (m < 49) {
                float v = fmaf(acc[j], sc1, sh1);
                v = fmaxf(v, 0.0f);
                if (b == 0) v *= mask1[nn * 49 + m];
                ipad[nn * 81 + (m / 7 + 1) * 9 + (m % 7) + 1] = v;
            }
        }
    };
    epi1(c1a, mt0);
    epi1(c1b, mt0 + 2);

    // ---------------- conv2 ---------------- (lambda starts with barrier)
    v8f c2a = {}, c2b = {};
    conv_pass(ipad, wt + 36864, c2a, c2b);
    __syncthreads();   // all waves done reading ipad before we overwrite it

    // Epilogue 2: mask2(batch 0) -> BN2 -> +residual(xpad) -> ReLU,
    // staged into ipad so the global store is coalesced.
    const float sc2 = g2[nn] * rsqrtf(v2p[nn] + EPSV);
    const float sh2 = b2[nn] - m2[nn] * sc2;
    auto epi2 = [&](const v8f& acc, int mt) {
        #pragma unroll
        for (int j = 0; j < 8; ++j) {
            int m = mt * 16 + j + 8 * hi;
            if (m < 49) {
                float v = acc[j];
                if (b == 0) v *= mask2[nn * 49 + m];
                v = fmaf(v, sc2, sh2);
                int off = nn * 81 + (m / 7 + 1) * 9 + (m % 7) + 1;
                v += xpad[off];
                ipad[off] = fmaxf(v, 0.0f);
            }
        }
    };
    epi2(c2a, mt0);
    epi2(c2b, mt0 + 2);
    __syncthreads();

    // Coalesced store of the whole image tile.
    float* ob = out + (size_t)b * 3136;
    for (int i = tid; i < 3136; i += 256) {
        int c = i / 49, m = i % 49;
        ob[i] = ipad[c * 81 + (m / 7 + 1) * 9 + (m % 7) + 1];
    }
}

// ---------------------------------------------------------------------------
extern "C" void kernel_launch(void* const* d_in, const int* in_sizes, int n_in,
                              void* d_out, int out_size, void* d_ws, size_t ws_size,
                              hipStream_t stream) {
    const float* x     = (const float*)d_in[0];
    const float* w1    = (const float*)d_in[1];
    const float* g1    = (const float*)d_in[2];
    const float* b1    = (const float*)d_in[3];
    const float* m1    = (const float*)d_in[4];
    const float* v1    = (const float*)d_in[5];
    const float* w2    = (const float*)d_in[6];
    const float* g2    = (const float*)d_in[7];
    const float* b2    = (const float*)d_in[8];
    const float* m2    = (const float*)d_in[9];
    const float* v2p   = (const float*)d_in[10];
    const float* mask1 = (const float*)d_in[11];
    const float* mask2 = (const float*)d_in[12];
    float* wt = (float*)d_ws;          // 2*36864 floats = 294,912 bytes
    float* out = (float*)d_out;

    prep_weights_kernel<<<288, 256, 0, stream>>>(w1, w2, wt);
    fused_basicblock_kernel<<<4096, 256, 0, stream>>>(
        x, wt, g1, b1, m1, v1, g2, b2, m2, v2p, mask1, mask2, out);
}